// PMLP_SGC_79353815761144
// MI455X (gfx1250) — compile-verified
//
#include <hip/hip_runtime.h>

// ---------------------------------------------------------------------------
// PMLP/SGC pipeline for MI455X (gfx1250, wave32).
//   5x SpMM (COO gather + f32 atomic scatter; 25.6MB working set lives in L2;
//            edge metadata scalarized to the SGPR/KMcnt path via readfirstlane)
//   GEMM1 (50000x128 @ 128x64) via V_WMMA_F32_16X16X4_F32 + fused BN stats
//   BN finalize (mean / rsqrt(var+eps))
//   GEMM2 (normalize+ReLU fused into A-operand) via V_WMMA_F32_16X16X4_F32
// ---------------------------------------------------------------------------

typedef __attribute__((ext_vector_type(2))) float v2f;
typedef __attribute__((ext_vector_type(8))) float v8f;

#define N_NODES 50000
#define C_IN    128
#define HID     64
#define C_OUT   40
#define NUM_MPS 5
#define BN_EPS  1e-10f

__global__ void zero4_kernel(float4* __restrict__ p, int n4) {
  int i = blockIdx.x * blockDim.x + threadIdx.x;
  if (i < n4) p[i] = make_float4(0.f, 0.f, 0.f, 0.f);
}

// One wave per edge; each lane handles 4 of the 128 features (b128 gather,
// 4x global_atomic_add_f32 scatter). Edge id forced uniform so row/col/weight
// loads can take the scalar path.
__global__ void spmm_kernel(const float* __restrict__ xin,
                            float* __restrict__ xout,
                            const int* __restrict__ eidx,
                            const float* __restrict__ ew,
                            int E) {
  int t    = blockIdx.x * blockDim.x + threadIdx.x;
  int lane = t & 31;
  int e    = __builtin_amdgcn_readfirstlane(t >> 5);   // wave-uniform edge id
  if (e >= E) return;
  int   row = eidx[e];        // edge_index[0][e]
  int   col = eidx[E + e];    // edge_index[1][e]
  float w   = ew[e];
  float4 v = *((const float4*)(xin + (size_t)col * C_IN) + lane);
  float* dst = xout + (size_t)row * C_IN + lane * 4;
  atomicAdd(dst + 0, w * v.x);
  atomicAdd(dst + 1, w * v.y);
  atomicAdd(dst + 2, w * v.z);
  atomicAdd(dst + 3, w * v.w);
}

// H = X @ W1^T + b1, plus per-channel sum / sumsq for BatchNorm.
// One wave computes one 16x16 output tile; K looped in steps of 4 (f32 WMMA).
__global__ void __launch_bounds__(256)
gemm1_kernel(const float* __restrict__ X,    // N x 128 (row major)
             const float* __restrict__ W1,   // 64 x 128 (row major) -> B = W1^T
             const float* __restrict__ b1,   // 64
             float* __restrict__ H,          // N x 64
             float* __restrict__ stats) {    // sum[64] | sumsq[64]
  int wave = (blockIdx.x * blockDim.x + threadIdx.x) >> 5;
  int lane = threadIdx.x & 31;
  const int MT = N_NODES / 16;   // 3125
  const int NT = HID / 16;       // 4
  if (wave >= MT * NT) return;
  int mt = wave >> 2;
  int nt = wave & 3;
  int m0 = mt * 16, n0 = nt * 16;
  int l15 = lane & 15;
  int hi  = lane >> 4;           // 0: K=k,k+1  1: K=k+2,k+3

  v8f acc = {};
  const float* arow = X  + (size_t)(m0 + l15) * C_IN + 2 * hi;
  const float* brow = W1 + (size_t)(n0 + l15) * C_IN + 2 * hi;
  #pragma unroll 8
  for (int k = 0; k < C_IN; k += 4) {
    v2f a = *(const v2f*)(arow + k);
    v2f b = *(const v2f*)(brow + k);
    acc = __builtin_amdgcn_wmma_f32_16x16x4_f32(
        /*neg_a=*/false, a, /*neg_b=*/false, b,
        /*c_mod=*/(short)0, acc, /*reuse_a=*/false, /*reuse_b=*/false);
  }

  int   n    = n0 + l15;                 // output channel (same for all 8 rows)
  float bias = b1[n];
  float s = 0.f, sq = 0.f;
  #pragma unroll
  for (int r = 0; r < 8; ++r) {
    int   m = m0 + r + 8 * hi;           // C/D layout: lanes 16-31 hold M=8..15
    float v = acc[r] + bias;
    H[(size_t)m * HID + n] = v;
    s += v;  sq += v * v;
  }
  atomicAdd(&stats[n],      s);
  atomicAdd(&stats[64 + n], sq);
}

__global__ void bn_finalize_kernel(float* __restrict__ stats) {
  int c = threadIdx.x;
  if (c < HID) {
    float inv_n = 1.0f / (float)N_NODES;
    float mean  = stats[c] * inv_n;
    float var   = stats[64 + c] * inv_n - mean * mean;
    stats[128 + c] = mean;
    stats[192 + c] = rsqrtf(var + BN_EPS);
  }
}

// OUT = relu((H - mean) * rstd) @ W2^T + b2.  Normalize+ReLU fused into the
// A operand per K-step; N padded to 48 with guarded (clamped) B loads/stores.
__global__ void __launch_bounds__(256)
gemm2_kernel(const float* __restrict__ H,     // N x 64
             const float* __restrict__ W2,    // 40 x 64 (row major) -> B = W2^T
             const float* __restrict__ b2,    // 40
             const float* __restrict__ stats, // mean @128, rstd @192
             float* __restrict__ OUT) {       // N x 40
  int wave = (blockIdx.x * blockDim.x + threadIdx.x) >> 5;
  int lane = threadIdx.x & 31;
  const int MT = N_NODES / 16;   // 3125
  const int NT = 3;              // ceil(40/16)
  if (wave >= MT * NT) return;
  int mt = wave / NT;
  int nt = wave - mt * NT;
  int m0 = mt * 16, n0 = nt * 16;
  int l15 = lane & 15;
  int hi  = lane >> 4;
  int n   = n0 + l15;
  int nn  = (n < C_OUT) ? n : (C_OUT - 1);   // clamp to stay in-bounds

  const float* mean = stats + 128;
  const float* rstd = stats + 192;

  v8f acc = {};
  const float* arow = H  + (size_t)(m0 + l15) * HID + 2 * hi;
  const float* brow = W2 + (size_t)nn * HID + 2 * hi;
  #pragma unroll 4
  for (int k = 0; k < HID; k += 4) {
    int kk = k + 2 * hi;
    v2f a = *(const v2f*)(arow + k);
    float a0 = (a.x - mean[kk])     * rstd[kk];
    float a1 = (a.y - mean[kk + 1]) * rstd[kk + 1];
    a.x = a0 > 0.f ? a0 : 0.f;
    a.y = a1 > 0.f ? a1 : 0.f;
    v2f b = *(const v2f*)(brow + k);
    if (n >= C_OUT) { b.x = 0.f; b.y = 0.f; }   // zero padded columns
    acc = __builtin_amdgcn_wmma_f32_16x16x4_f32(
        false, a, false, b, (short)0, acc, false, false);
  }

  if (n < C_OUT) {
    float bias = b2[n];
    #pragma unroll
    for (int r = 0; r < 8; ++r) {
      int m = m0 + r + 8 * hi;
      OUT[(size_t)m * C_OUT + n] = acc[r] + bias;
    }
  }
}

extern "C" void kernel_launch(void* const* d_in, const int* in_sizes, int n_in,
                              void* d_out, int out_size, void* d_ws, size_t ws_size,
                              hipStream_t stream) {
  const float* x    = (const float*)d_in[0];
  const int*   eidx = (const int*)  d_in[1];   // (2, E) row major
  const float* ew   = (const float*)d_in[2];
  const float* W1   = (const float*)d_in[3];
  const float* b1   = (const float*)d_in[4];
  const float* W2   = (const float*)d_in[5];
  const float* b2   = (const float*)d_in[6];
  float* out = (float*)d_out;
  int E = in_sizes[2];                         // edge_weight element count

  // Workspace layout (floats):
  float* bufA  = (float*)d_ws;                          // N*128
  float* bufB  = bufA + (size_t)N_NODES * C_IN;         // N*128
  float* Hbuf  = bufB + (size_t)N_NODES * C_IN;         // N*64
  float* stats = Hbuf + (size_t)N_NODES * HID;          // 256 floats

  dim3 blk(256);
  const int nfeat  = N_NODES * C_IN;
  const int nfeat4 = nfeat / 4;

  // 5-hop SGC propagation, ping-pong between bufA/bufB.
  const float* cur_in  = x;
  float*       cur_out = bufA;
  for (int r = 0; r < NUM_MPS; ++r) {
    zero4_kernel<<<(nfeat4 + 255) / 256, blk, 0, stream>>>((float4*)cur_out, nfeat4);
    int threads = E * 32;   // one wave per edge
    spmm_kernel<<<(threads + 255) / 256, blk, 0, stream>>>(cur_in, cur_out, eidx, ew, E);
    cur_in  = cur_out;
    cur_out = (cur_out == bufA) ? bufB : bufA;
  }
  // After 5 rounds cur_in == bufA.

  zero4_kernel<<<1, blk, 0, stream>>>((float4*)stats, 32);   // 128 floats
  {
    int waves  = (N_NODES / 16) * (HID / 16);   // 12500 tiles
    int blocks = (waves * 32 + 255) / 256;
    gemm1_kernel<<<blocks, blk, 0, stream>>>(cur_in, W1, b1, Hbuf, stats);
  }
  bn_finalize_kernel<<<1, 64, 0, stream>>>(stats);
  {
    int waves  = (N_NODES / 16) * 3;            // 9375 tiles (N padded to 48)
    int blocks = (waves * 32 + 255) / 256;
    gemm2_kernel<<<blocks, blk, 0, stream>>>(Hbuf, W2, b2, stats, out);
  }
}